// SConv2d_v0_13065290515014
// MI455X (gfx1250) — compile-verified
//
#include <hip/hip_runtime.h>

// SConv2d "nonlinear tree conv": out[n,o,h,w] = tree27( x_nbhd * W[o] ) + bias[o]
// tree combiner f(a,b,c) = (a+b+c - a*b*c)*0.5, applied 27->9->3->1.
// N=8, C=3, H=W=64, O=64, K=3, P=1.
//
// Design (MI455X, wave32):
//  - Two pixels per lane (images n and n+4, same (h,w)) packed in an
//    ext_vector float2 -> VOP3P packed-fp32 (v_pk_mul/add/fma) VALU path.
//  - Wave-invariant weights/bias read via scalar loads (constant cache).
//  - Branch-free zero padding: clamped addresses + v_cndmask selects.
//  - NT stores for the 8 MB streamed output; x (393 KB) stays L2-resident.

typedef __attribute__((ext_vector_type(2))) float vf2;

#define DEVFN static __device__ __forceinline__

DEVFN vf2 comb(vf2 a, vf2 b, vf2 c) {
    vf2 t = a + b + c;
    vf2 u = a * b;
    return (t - u * c) * 0.5f;   // t - u*c contracts to v_pk_fma(-u, c, t)
}

__global__ __launch_bounds__(256) void sconv2d_tree_kernel(
    const float* __restrict__ x,     // [8,3,64,64]
    const float* __restrict__ wgt,   // [64,3,3,3] == [64,27]
    const float* __restrict__ bias,  // [64]
    float* __restrict__ out)         // [8,64,64,64]
{
    // Each thread owns one (h,w) position in two images: n0 (0..3) and n0+4.
    const int t  = blockIdx.x * 256 + threadIdx.x;   // 0..16383
    const int w0 = t & 63;
    const int h0 = (t >> 6) & 63;
    const int n0 = t >> 12;                          // 0..3

    const float* xA = x + n0 * (3 * 64 * 64);        // image n0
    const float* xB = xA + 4 * (3 * 64 * 64);        // image n0+4

    // Load the 3x3x3 padded neighborhood for both pixels into registers.
    // Branch-free: clamp address into bounds, select 0 on OOB (v_cndmask),
    // so all 54 loads are unconditional and coalesced.
    vf2 xv[27];
#pragma unroll
    for (int dh = 0; dh < 3; ++dh) {
        const int  hraw = h0 + dh - 1;
        const bool hok  = (unsigned)hraw < 64u;
        const int  hh   = hok ? hraw : 0;
#pragma unroll
        for (int dw = 0; dw < 3; ++dw) {
            const int  wraw = w0 + dw - 1;
            const bool wok  = (unsigned)wraw < 64u;
            const int  ww   = wok ? wraw : 0;
            const bool ok   = hok && wok;
            const int  off  = hh * 64 + ww;
#pragma unroll
            for (int c = 0; c < 3; ++c) {
                const float a = xA[c * 4096 + off];
                const float b = xB[c * 4096 + off];
                vf2 v;
                v.x = ok ? a : 0.0f;
                v.y = ok ? b : 0.0f;
                xv[c * 9 + dh * 3 + dw] = v;   // l = c*9 + dh*3 + dw (ref order)
            }
        }
    }

    // Split the O dimension across gridDim.y: each block does 8 output channels.
    const int o_beg = blockIdx.y * 8;
    const int o_end = o_beg + 8;

    // Output base indices for the two pixels.
    const int pixoff = h0 * 64 + w0;
    const int baseA  = n0 * (64 * 4096) + pixoff;        // + o*4096
    const int baseB  = baseA + 4 * (64 * 4096);

#pragma unroll 2
    for (int o = o_beg; o < o_end; ++o) {
        const float* wo = wgt + o * 27;
        // Prefetch next channel's weights (speculative; dropped if OOB).
        __builtin_prefetch(wo + 27, 0, 0);

        // Uniform (wave-invariant) weight reads -> s_load / SGPR broadcast.
        float ws[27];
#pragma unroll
        for (int i = 0; i < 27; ++i) ws[i] = wo[i];

        // Level 1: 27 products -> 9 ; Level 2: 9 -> 3 ; Level 3: 3 -> 1.
        vf2 r[3];
#pragma unroll
        for (int c = 0; c < 3; ++c) {
            vf2 q[3];
#pragma unroll
            for (int dh = 0; dh < 3; ++dh) {
                const int l = c * 9 + dh * 3;
                vf2 p0 = xv[l + 0] * ws[l + 0];
                vf2 p1 = xv[l + 1] * ws[l + 1];
                vf2 p2 = xv[l + 2] * ws[l + 2];
                q[dh] = comb(p0, p1, p2);
            }
            r[c] = comb(q[0], q[1], q[2]);
        }
        vf2 s = comb(r[0], r[1], r[2]);

        const float bo = bias[o];
        // Output is 8 MB streamed, never re-read: non-temporal stores keep
        // L2 free for x / weights.
        __builtin_nontemporal_store(s.x + bo, &out[baseA + o * 4096]);
        __builtin_nontemporal_store(s.y + bo, &out[baseB + o * 4096]);
    }
}

extern "C" void kernel_launch(void* const* d_in, const int* in_sizes, int n_in,
                              void* d_out, int out_size, void* d_ws, size_t ws_size,
                              hipStream_t stream) {
    (void)in_sizes; (void)n_in; (void)out_size; (void)d_ws; (void)ws_size;
    const float* x    = (const float*)d_in[0];
    const float* wgt  = (const float*)d_in[1];
    const float* bias = (const float*)d_in[2];
    float* out        = (float*)d_out;

    dim3 grid(64, 8);   // 64 blocks cover 16384 pixel-pairs; y splits O into 8x8
    dim3 block(256);
    sconv2d_tree_kernel<<<grid, block, 0, stream>>>(x, wgt, bias, out);
}